// Embedding_86260123172998
// MI455X (gfx1250) — compile-verified
//
#include <hip/hip_runtime.h>
#include <math.h>

typedef float v2f __attribute__((ext_vector_type(2)));
typedef float v8f __attribute__((ext_vector_type(8)));

#define NB 8
#define NS 2048
#define ND 3
#define NF 125
#define FP 128        // padded feature dim
#define KPAD (FP + 4) // transposed-row stride: (4n + k) mod 64 banks, conflict-free
#define NK 16
#define NPTS (NB*NS)

__device__ __forceinline__ float gelu_exact(float v) {
    return 0.5f * v * (1.0f + erff(v * 0.7071067811865475f));
}

// ---------------------------------------------------------------------------
// Kernel 1: res = gelu(x@W1 + b1) @ W2 + b2   (padded to FP=128 columns)
// One wave per 16-row M-tile. A-operand built directly in the
// V_WMMA_F32_16X16X4_F32 register layout; W2 staged TRANSPOSED in LDS so the
// B-operand pair is one aligned ds_load_b64 into an adjacent VGPR pair.
// ---------------------------------------------------------------------------
__global__ __launch_bounds__(256) void mlp_wmma_kernel(
    const float* __restrict__ x, const float* __restrict__ W1,
    const float* __restrict__ b1, const float* __restrict__ W2,
    const float* __restrict__ b2, float* __restrict__ resw)
{
    __shared__ float sW2t[FP * KPAD];   // 66 KB: sW2t[n][k] = W2[k][n] (zero-padded)
    const int tid = threadIdx.x;
    for (int i = tid; i < FP*FP; i += 256) {
        const int k = i >> 7, n = i & 127;          // n fastest -> coalesced W2 reads
        sW2t[n*KPAD + k] = (k < NF && n < NF) ? W2[k*NF + n] : 0.0f;
    }
    __syncthreads();

    const int lane = tid & 31;
    const int wave = tid >> 5;
    const int row  = lane & 15;    // M row within tile (and N col for B/C/D)
    const int half = lane >> 4;    // 0: K pair {4k,4k+1}, 1: K pair {4k+2,4k+3}
    const int m0   = (blockIdx.x * 8 + wave) * 16;

    const float px = x[(size_t)(m0+row)*3 + 0];
    const float py = x[(size_t)(m0+row)*3 + 1];
    const float pz = x[(size_t)(m0+row)*3 + 2];

    // A-operand: h[row, c] for c = 4*ks + 2*half + u; every tile element is
    // computed exactly once across the wave. K padded 125 -> 128 with zeros.
    float areg[64];
    #pragma unroll
    for (int ks = 0; ks < 32; ++ks) {
        #pragma unroll
        for (int u = 0; u < 2; ++u) {
            const int c = 4*ks + 2*half + u;
            float v = 0.0f;
            if (c < NF) {
                v = px*W1[0*NF + c] + py*W1[1*NF + c] + pz*W1[2*NF + c] + b1[c];
                v = gelu_exact(v);
            }
            areg[2*ks + u] = v;
        }
    }

    for (int nt = 0; nt < 8; ++nt) {
        const int n = nt*16 + row;
        const float bias = (n < NF) ? b2[n] : 0.0f;
        v8f acc;
        #pragma unroll
        for (int r = 0; r < 8; ++r) acc[r] = bias;   // C init: bias per column

        const float* bn = &sW2t[n * KPAD];           // this lane's B column
        #pragma unroll
        for (int ks = 0; ks < 32; ++ks) {
            const int kb = 4*ks + 2*half;            // even -> 8B aligned
            v2f a, b;
            a.x = areg[2*ks + 0];
            a.y = areg[2*ks + 1];
            b   = *(const v2f*)(bn + kb);            // single ds_load_b64
            acc = __builtin_amdgcn_wmma_f32_16x16x4_f32(
                    false, a, false, b, (short)0, acc, false, false);
        }

        // D layout: VGPR r -> M = half*8 + r, N = row
        #pragma unroll
        for (int r = 0; r < 8; ++r) {
            const int m = half*8 + r;
            resw[(size_t)(m0 + m)*FP + nt*16 + row] = acc[r];
        }
    }
}

// ---------------------------------------------------------------------------
// Kernel 2: brute-force 16-NN in 3D per batch. One thread per query point,
// whole batch (2048 pts + norms, 32KB) staged in LDS; sorted top-16 insertion
// in registers (stable ties like lax.top_k).
// ---------------------------------------------------------------------------
__global__ __launch_bounds__(256) void knn_kernel(
    const float* __restrict__ x, int* __restrict__ knn)
{
    __shared__ float spx[NS], spy[NS], spz[NS], ssq[NS];   // 32 KB
    const int b  = blockIdx.x >> 3;          // 8 blocks per batch
    const int i0 = (blockIdx.x & 7) * 256;
    const float* xb = x + (size_t)b*NS*3;
    for (int j = threadIdx.x; j < NS; j += 256) {
        const float a0 = xb[j*3+0], a1 = xb[j*3+1], a2 = xb[j*3+2];
        spx[j] = a0; spy[j] = a1; spz[j] = a2;
        ssq[j] = a0*a0 + a1*a1 + a2*a2;
    }
    __syncthreads();

    const int i = i0 + threadIdx.x;
    const float qx = spx[i], qy = spy[i], qz = spz[i], qs = ssq[i];

    float dist[NK]; int idx[NK];
    #pragma unroll
    for (int k = 0; k < NK; ++k) { dist[k] = 3.4e38f; idx[k] = 0; }

    for (int j = 0; j < NS; ++j) {
        const float d = qs + ssq[j] - 2.0f*(qx*spx[j] + qy*spy[j] + qz*spz[j]);
        if (d < dist[NK-1]) {
            int pos = NK-1;
            #pragma unroll
            for (int t = NK-1; t > 0; --t) {
                if (dist[t-1] > d) {           // strict >: stable on ties
                    dist[t] = dist[t-1]; idx[t] = idx[t-1]; pos = t-1;
                }
            }
            dist[pos] = d; idx[pos] = j;
        }
    }

    int* kout = knn + (size_t)(b*NS + i)*NK;
    #pragma unroll
    for (int k = 0; k < NK; ++k) kout[k] = idx[k];
}

// ---------------------------------------------------------------------------
// Kernel 3: out[q, 0:125] = max_k res[b, knn[q,k], :];  out[q,125:128] = x[q]
// 128 threads per query point (one per output column); res is L2-resident.
// ---------------------------------------------------------------------------
__global__ __launch_bounds__(256) void gather_max_kernel(
    const float* __restrict__ x, const float* __restrict__ resw,
    const int* __restrict__ knn, float* __restrict__ out)
{
    __shared__ int sidx[2][NK];
    const int sub = threadIdx.x >> 7;        // which of the 2 queries
    const int d   = threadIdx.x & 127;       // output column
    const int q   = blockIdx.x * 2 + sub;    // flat query index 0..16383
    if (d < NK) sidx[sub][d] = knn[(size_t)q*NK + d];
    __syncthreads();

    if (d < NF) {
        const int b = q >> 11;
        const float* rb = resw + (size_t)b*NS*FP;
        float m = -3.4e38f;
        #pragma unroll
        for (int k = 0; k < NK; ++k)
            m = fmaxf(m, rb[(size_t)sidx[sub][k]*FP + d]);
        out[(size_t)q*FP + d] = m;
    } else {
        out[(size_t)q*FP + d] = x[(size_t)q*3 + (d - NF)];
    }
}

// ---------------------------------------------------------------------------
extern "C" void kernel_launch(void* const* d_in, const int* in_sizes, int n_in,
                              void* d_out, int out_size, void* d_ws, size_t ws_size,
                              hipStream_t stream) {
    const float* x  = (const float*)d_in[0];
    const float* W1 = (const float*)d_in[1];
    const float* b1 = (const float*)d_in[2];
    const float* W2 = (const float*)d_in[3];
    const float* b2 = (const float*)d_in[4];
    float* out = (float*)d_out;

    float* resw = (float*)d_ws;                                        // 16384*128 f32 = 8 MB
    int*   knn  = (int*)((char*)d_ws + (size_t)NPTS*FP*sizeof(float)); // 16384*16 i32 = 1 MB

    mlp_wmma_kernel<<<NPTS/(16*8), 256, 0, stream>>>(x, W1, b1, W2, b2, resw);
    knn_kernel<<<NB*(NS/256), 256, 0, stream>>>(x, knn);
    gather_max_kernel<<<NPTS/2, 256, 0, stream>>>(x, resw, knn, out);
}